// GAT_45621142618169
// MI455X (gfx1250) — compile-verified
//
#include <hip/hip_runtime.h>
#include <hip/hip_bf16.h>
#include <math.h>

typedef __attribute__((ext_vector_type(2))) float v2f;
typedef __attribute__((ext_vector_type(8))) float v8f;

#define THREADS 256

static inline int cdiv_ll(long long a, int b) { return (int)((a + b - 1) / b); }

// float atomic max via monotone int-bit trick (sentinel = 0xFFFFFFFF memset).
__device__ __forceinline__ void atomicMaxF(float* addr, float val) {
  if (val >= 0.0f) atomicMax((int*)addr, __float_as_int(val));
  else             atomicMin((unsigned int*)addr, __float_as_uint(val));
}

__device__ __forceinline__ float lrelu02(float v) { return v > 0.0f ? v : 0.2f * v; }

// ---------------------------------------------------------------------------
// D = A[M,K] @ B[K,Nc] in fp32 via V_WMMA_F32_16X16X4_F32.
// One wave per 16x16 output tile. Requires M%16==0, K%4==0, Nc%16==0.
// A-frag (ISA 16x4 layout): lanes 0-15 hold (M=l16, K=k..k+1), lanes 16-31
// hold (M=l16, K=k+2..k+3) -> contiguous float2 per lane.
// C/D: VGPR i = row (half*8 + i), col = l16.
// ---------------------------------------------------------------------------
__global__ void gemm_wmma_f32(const float* __restrict__ A,
                              const float* __restrict__ B,
                              float* __restrict__ C,
                              int M, int K, int Nc) {
  const int tilesN = Nc >> 4;
  const int totalTiles = (M >> 4) * tilesN;
  const int wave = (int)((blockIdx.x * blockDim.x + threadIdx.x) >> 5);
  if (wave >= totalTiles) return;            // wave-uniform: EXEC all-1s at WMMA
  const int mt = wave / tilesN;
  const int nt = wave - mt * tilesN;
  const int lane = threadIdx.x & 31;
  const int half = lane >> 4;
  const int l16  = lane & 15;

  const float* __restrict__ Arow = A + (size_t)(mt * 16 + l16) * K;
  const float* __restrict__ Bcol = B + (nt * 16 + l16);

  v8f acc = {};
  for (int k = 0; k < K; k += 4) {
    const int ka = k + half * 2;
    v2f a, b;
    a.x = Arow[ka];
    a.y = Arow[ka + 1];
    b.x = Bcol[(size_t)ka * Nc];
    b.y = Bcol[(size_t)(ka + 1) * Nc];
    acc = __builtin_amdgcn_wmma_f32_16x16x4_f32(false, a, false, b,
                                                (short)0, acc, false, false);
  }
  float* Cp = C + (size_t)(mt * 16 + half * 8) * Nc + nt * 16 + l16;
#pragma unroll
  for (int i = 0; i < 8; ++i) Cp[(size_t)i * Nc] = acc[i];
}

// es[n,h] = <h[n,h,:], a_src[h,:]>, ed likewise
__global__ void attn_scores(const float* __restrict__ h,
                            const float* __restrict__ as_,
                            const float* __restrict__ ad_,
                            float* __restrict__ es, float* __restrict__ ed,
                            int N, int H, int D) {
  int idx = blockIdx.x * blockDim.x + threadIdx.x;
  if (idx >= N * H) return;
  int n = idx / H, hh = idx - n * H;
  const float* hp = h + (size_t)n * H * D + (size_t)hh * D;
  const float* ap = as_ + hh * D;
  const float* bp = ad_ + hh * D;
  float s = 0.f, d = 0.f;
  for (int i = 0; i < D; ++i) {
    float v = hp[i];
    s = fmaf(v, ap[i], s);
    d = fmaf(v, bp[i], d);
  }
  es[idx] = s;
  ed[idx] = d;
}

// Pass A: per-dst running max of leaky_relu(es[src]+ed[dst])
__global__ void edge_max(const int* __restrict__ src, const int* __restrict__ dst,
                         int E, int N,
                         const float* __restrict__ es, const float* __restrict__ ed,
                         float* __restrict__ m, int H) {
  int e = blockIdx.x * blockDim.x + threadIdx.x;
  int Etot = E + N;
  if (e >= Etot) return;
  int s = (e < E) ? src[e] : (e - E);   // self-loops appended
  int d = (e < E) ? dst[e] : (e - E);
  for (int hh = 0; hh < H; ++hh) {
    float v = lrelu02(es[s * H + hh] + ed[d * H + hh]);
    atomicMaxF(&m[d * H + hh], v);
  }
}

// Pass B: p = exp(e - m[dst]); cache p, accumulate per-dst sum
__global__ void edge_expsum(const int* __restrict__ src, const int* __restrict__ dst,
                            int E, int N,
                            const float* __restrict__ es, const float* __restrict__ ed,
                            const float* __restrict__ m, float* __restrict__ ss,
                            float* __restrict__ p, int H) {
  int e = blockIdx.x * blockDim.x + threadIdx.x;
  int Etot = E + N;
  if (e >= Etot) return;
  int s = (e < E) ? src[e] : (e - E);
  int d = (e < E) ? dst[e] : (e - E);
  for (int hh = 0; hh < H; ++hh) {
    float v  = lrelu02(es[s * H + hh] + ed[d * H + hh]);
    float pe = expf(v - m[d * H + hh]);
    p[(size_t)e * H + hh] = pe;
    atomicAdd(&ss[d * H + hh], pe);
  }
}

// Pass C: out[dst,:] += alpha * h[src,:]  (one wave per edge, lane = channel)
__global__ void edge_aggregate(const int* __restrict__ src, const int* __restrict__ dst,
                               int E, int N,
                               const float* __restrict__ h, const float* __restrict__ p,
                               const float* __restrict__ ss, float* __restrict__ out,
                               int H, int D) {
  int gid = blockIdx.x * blockDim.x + threadIdx.x;
  int wave = gid >> 5;
  int lane = gid & 31;
  int Etot = E + N;
  if (wave >= Etot) return;
  int s = (wave < E) ? src[wave] : (wave - E);
  int d = (wave < E) ? dst[wave] : (wave - E);
  int C = H * D;
  for (int c = lane; c < C; c += 32) {
    int hh = c / D;
    float alpha = p[(size_t)wave * H + hh] / (ss[d * H + hh] + 1e-16f);
    atomicAdd(&out[(size_t)d * C + c], h[(size_t)s * C + c] * alpha);
  }
}

// In-place: x = elu(x + b[c])
__global__ void bias_elu(float* __restrict__ x, const float* __restrict__ b,
                         int N, int C) {
  int i = blockIdx.x * blockDim.x + threadIdx.x;
  if (i >= N * C) return;
  float v = x[i] + b[i % C];
  x[i] = v > 0.f ? v : expm1f(v);
}

// Layer 3 tail: elu(agg + b) then row-wise log_softmax over C (<=16) classes
__global__ void final_elu_logsoftmax(const float* __restrict__ agg,
                                     const float* __restrict__ b,
                                     float* __restrict__ out, int N, int C) {
  int n = blockIdx.x * blockDim.x + threadIdx.x;
  if (n >= N) return;
  float v[16];
  float mx = -INFINITY;
  for (int c = 0; c < C; ++c) {
    float t = agg[(size_t)n * C + c] + b[c];
    t = t > 0.f ? t : expm1f(t);
    v[c] = t;
    mx = fmaxf(mx, t);
  }
  float s = 0.f;
  for (int c = 0; c < C; ++c) s += expf(v[c] - mx);
  float lse = mx + logf(s);
  for (int c = 0; c < C; ++c) out[(size_t)n * C + c] = v[c] - lse;
}

extern "C" void kernel_launch(void* const* d_in, const int* in_sizes, int n_in,
                              void* d_out, int out_size, void* d_ws, size_t ws_size,
                              hipStream_t stream) {
  const float* x   = (const float*)d_in[0];
  const float* W1  = (const float*)d_in[1];
  const float* a1s = (const float*)d_in[2];
  const float* a1d = (const float*)d_in[3];
  const float* b1  = (const float*)d_in[4];
  const float* W2  = (const float*)d_in[5];
  const float* a2s = (const float*)d_in[6];
  const float* a2d = (const float*)d_in[7];
  const float* b2  = (const float*)d_in[8];
  const float* W3  = (const float*)d_in[9];
  const float* a3s = (const float*)d_in[10];
  const float* a3d = (const float*)d_in[11];
  const float* b3  = (const float*)d_in[12];
  const int*   ei  = (const int*)d_in[13];

  const int HD = in_sizes[4];        // 96 = heads*dim
  const int F  = in_sizes[1] / HD;   // 128
  const int N  = in_sizes[0] / F;    // 100000
  const int C3 = in_sizes[12];       // 16
  const int E  = in_sizes[13] / 2;   // 1600000
  const int H  = 3;
  const int D  = HD / H;             // 32
  const int Etot = E + N;

  const int* src = ei;
  const int* dst = ei + E;

  float* ws   = (float*)d_ws;
  float* bufH = ws;                              // [N, HD]  h = in @ W
  float* bufO = bufH + (size_t)N * HD;           // [N, HD]  aggregated / next input
  float* es   = bufO + (size_t)N * HD;           // [N, H]
  float* ed   = es + (size_t)N * H;              // [N, H]
  float* m    = ed + (size_t)N * H;              // [N, H]
  float* ss   = m  + (size_t)N * H;              // [N, H]
  float* p    = ss + (size_t)N * H;              // [Etot, H]

  // ---------------- Layer 1 (in = x, F=128 -> HD=96) ----------------
  {
    int tiles = (N / 16) * (HD / 16);
    gemm_wmma_f32<<<cdiv_ll((long long)tiles * 32, THREADS), THREADS, 0, stream>>>(
        x, W1, bufH, N, F, HD);
    attn_scores<<<cdiv_ll((long long)N * H, THREADS), THREADS, 0, stream>>>(
        bufH, a1s, a1d, es, ed, N, H, D);
    hipMemsetAsync(m, 0xFF, (size_t)N * H * sizeof(float), stream);
    hipMemsetAsync(ss, 0, (size_t)N * H * sizeof(float), stream);
    hipMemsetAsync(bufO, 0, (size_t)N * HD * sizeof(float), stream);
    edge_max<<<cdiv_ll(Etot, THREADS), THREADS, 0, stream>>>(src, dst, E, N, es, ed, m, H);
    edge_expsum<<<cdiv_ll(Etot, THREADS), THREADS, 0, stream>>>(src, dst, E, N, es, ed, m, ss, p, H);
    edge_aggregate<<<cdiv_ll((long long)Etot * 32, THREADS), THREADS, 0, stream>>>(
        src, dst, E, N, bufH, p, ss, bufO, H, D);
    bias_elu<<<cdiv_ll((long long)N * HD, THREADS), THREADS, 0, stream>>>(bufO, b1, N, HD);
  }

  // ---------------- Layer 2 (in = bufO, 96 -> 96) ----------------
  {
    int tiles = (N / 16) * (HD / 16);
    gemm_wmma_f32<<<cdiv_ll((long long)tiles * 32, THREADS), THREADS, 0, stream>>>(
        bufO, W2, bufH, N, HD, HD);
    attn_scores<<<cdiv_ll((long long)N * H, THREADS), THREADS, 0, stream>>>(
        bufH, a2s, a2d, es, ed, N, H, D);
    hipMemsetAsync(m, 0xFF, (size_t)N * H * sizeof(float), stream);
    hipMemsetAsync(ss, 0, (size_t)N * H * sizeof(float), stream);
    hipMemsetAsync(bufO, 0, (size_t)N * HD * sizeof(float), stream);  // after gemm read
    edge_max<<<cdiv_ll(Etot, THREADS), THREADS, 0, stream>>>(src, dst, E, N, es, ed, m, H);
    edge_expsum<<<cdiv_ll(Etot, THREADS), THREADS, 0, stream>>>(src, dst, E, N, es, ed, m, ss, p, H);
    edge_aggregate<<<cdiv_ll((long long)Etot * 32, THREADS), THREADS, 0, stream>>>(
        src, dst, E, N, bufH, p, ss, bufO, H, D);
    bias_elu<<<cdiv_ll((long long)N * HD, THREADS), THREADS, 0, stream>>>(bufO, b2, N, HD);
  }

  // ---------------- Layer 3 (in = bufO, 96 -> 16, 1 head) ----------------
  {
    int tiles = (N / 16) * (C3 / 16);
    gemm_wmma_f32<<<cdiv_ll((long long)tiles * 32, THREADS), THREADS, 0, stream>>>(
        bufO, W3, bufH, N, HD, C3);
    attn_scores<<<cdiv_ll(N, THREADS), THREADS, 0, stream>>>(
        bufH, a3s, a3d, es, ed, N, 1, C3);
    hipMemsetAsync(m, 0xFF, (size_t)N * sizeof(float), stream);
    hipMemsetAsync(ss, 0, (size_t)N * sizeof(float), stream);
    hipMemsetAsync(bufO, 0, (size_t)N * C3 * sizeof(float), stream);  // after gemm read
    edge_max<<<cdiv_ll(Etot, THREADS), THREADS, 0, stream>>>(src, dst, E, N, es, ed, m, 1);
    edge_expsum<<<cdiv_ll(Etot, THREADS), THREADS, 0, stream>>>(src, dst, E, N, es, ed, m, ss, p, 1);
    edge_aggregate<<<cdiv_ll((long long)Etot * 32, THREADS), THREADS, 0, stream>>>(
        src, dst, E, N, bufH, p, ss, bufO, 1, C3);
    final_elu_logsoftmax<<<cdiv_ll(N, THREADS), THREADS, 0, stream>>>(
        bufO, b3, (float*)d_out, N, C3);
  }
}